// GNNCritic_14516989461161
// MI455X (gfx1250) — compile-verified
//
#include <hip/hip_runtime.h>
#include <hip/hip_bf16.h>

typedef __attribute__((ext_vector_type(2))) float v2f;
typedef __attribute__((ext_vector_type(8))) float v8f;

// ---------------------------------------------------------------------------
// Utility: zero a float buffer
// ---------------------------------------------------------------------------
__global__ void zero_f(float* __restrict__ p, int n) {
    int i = blockIdx.x * blockDim.x + threadIdx.x;
    if (i < n) p[i] = 0.0f;
}

// ---------------------------------------------------------------------------
// deg[col] += ew[e]   (segment_sum over destination nodes)
// ---------------------------------------------------------------------------
__global__ void deg_accum(const int* __restrict__ ei, const float* __restrict__ ew,
                          float* __restrict__ deg, int E) {
    int e = blockIdx.x * blockDim.x + threadIdx.x;
    if (e < E) atomicAdd(&deg[ei[E + e]], ew[e]);
}

// dis[n] = rsqrt(deg[n] + 1)   (self-loop weight 1), in place
__global__ void dis_kernel(float* __restrict__ d, int N) {
    int n = blockIdx.x * blockDim.x + threadIdx.x;
    if (n < N) d[n] = rsqrtf(d[n] + 1.0f);
}

// ---------------------------------------------------------------------------
// WMMA fp32 GEMM:  out[N x 64] = A[N x 64] @ W[64 x 64]   (+ attention epilogue)
// Uses V_WMMA_F32_16X16X4_F32. One wave computes a 16x16 tile; block = 4 waves
// covering the 64 output columns; grid.x = N/16 row tiles.
//
// Layouts (ISA 7.12.2):
//   A 16x4 f32: lanes 0-15 row=lane K={k,k+1}; lanes 16-31 row=lane-16 K={k+2,k+3}
//   B 4x16 f32: VGPR0 = rows {k, k+2} striped lo/hi half-wave, VGPR1 = {k+1, k+3}
//   C/D 16x16 : VGPR i = rows {i, i+8}, N = lane&15
//
// mode 0: plain store
// mode 1: out = leaky_relu(acc + rowAdd[batch[row]*64 + col], 0.2)
// ---------------------------------------------------------------------------
__global__ void gemm64_wmma(const float* __restrict__ A, const float* __restrict__ W,
                            const float* __restrict__ rowAdd, const int* __restrict__ batch,
                            float* __restrict__ out, int mode) {
    const int wave = threadIdx.x >> 5;
    const int lane = threadIdx.x & 31;
    const int row0 = blockIdx.x * 16;
    const int col0 = wave * 16;
    const int r    = lane & 15;
    const int koff = (lane >> 4) << 1;       // 0 for lanes 0-15, 2 for lanes 16-31
    const int col  = col0 + r;

    const float* arow = A + (size_t)(row0 + r) * 64;

    v8f c = {0.f, 0.f, 0.f, 0.f, 0.f, 0.f, 0.f, 0.f};
#pragma unroll
    for (int k = 0; k < 64; k += 4) {
        v2f a, b;
        a.x = arow[k + koff];
        a.y = arow[k + koff + 1];
        b.x = W[(size_t)(k + koff)     * 64 + col];
        b.y = W[(size_t)(k + koff + 1) * 64 + col];
        c = __builtin_amdgcn_wmma_f32_16x16x4_f32(false, a, false, b,
                                                  (short)0, c, false, false);
    }

    const int rbase = row0 + ((lane >> 4) << 3);   // +0 or +8
#pragma unroll
    for (int i = 0; i < 8; ++i) {
        const int row = rbase + i;
        float v = c[i];
        if (mode == 1) {
            v += rowAdd[batch[row] * 64 + col];
            v = (v > 0.0f) ? v : 0.2f * v;         // leaky_relu(0.2)
        }
        out[(size_t)row * 64 + col] = v;
    }
}

// ---------------------------------------------------------------------------
// Edge scatter: agg[col] += dis[row]*ew*dis[col] * xw[row]
// 16 threads per edge, float4 gather, 4 global f32 atomics per thread.
// ---------------------------------------------------------------------------
__global__ void edge_scatter(const int* __restrict__ ei, const float* __restrict__ ew,
                             const float* __restrict__ dis, const float* __restrict__ xw,
                             float* __restrict__ agg, int E) {
    int t = blockIdx.x * blockDim.x + threadIdx.x;
    int e = t >> 4;
    if (e >= E) return;
    int fg  = (t & 15) << 2;
    int row = ei[e];
    int dst = ei[E + e];
    float norm = dis[row] * ew[e] * dis[dst];
    const float4 v = *(const float4*)(xw + (size_t)row * 64 + fg);
    float* p = agg + (size_t)dst * 64 + fg;
    atomicAdd(p + 0, norm * v.x);
    atomicAdd(p + 1, norm * v.y);
    atomicAdd(p + 2, norm * v.z);
    atomicAdd(p + 3, norm * v.w);
}

// h = relu(agg + dis^2 * xw + b)
__global__ void combine_relu(const float* __restrict__ agg, const float* __restrict__ xw,
                             const float* __restrict__ dis, const float* __restrict__ b,
                             float* __restrict__ h, int total) {
    int idx = blockIdx.x * blockDim.x + threadIdx.x;
    if (idx >= total) return;
    int n = idx >> 6, f = idx & 63;
    float d = dis[n];
    float v = agg[idx] + d * d * xw[idx] + b[f];
    h[idx] = (v > 0.0f) ? v : 0.0f;
}

// embW[b][c] = sum_k app[b][k] * Wa[64+k][c] + ba[c]   (8x64, single block)
__global__ void embw_kernel(const float* __restrict__ app, const float* __restrict__ Wa,
                            const float* __restrict__ ba, float* __restrict__ embW) {
    int t = threadIdx.x;              // 512 threads
    int bb = t >> 6, c = t & 63;
    float acc = ba[c];
#pragma unroll 4
    for (int k = 0; k < 64; ++k)
        acc += app[bb * 64 + k] * Wa[(size_t)(64 + k) * 64 + c];
    embW[t] = acc;
}

// s[n] = att[n] . Ws + bs
__global__ void scores_kernel(const float* __restrict__ att, const float* __restrict__ Ws,
                              const float* __restrict__ bs, float* __restrict__ s, int N) {
    int n = blockIdx.x * blockDim.x + threadIdx.x;
    if (n >= N) return;
    const float4* a = (const float4*)(att + (size_t)n * 64);
    const float4* w = (const float4*)Ws;
    float acc = 0.0f;
#pragma unroll
    for (int i = 0; i < 16; ++i) {
        float4 av = a[i], wv = w[i];
        acc += av.x * wv.x + av.y * wv.y + av.z * wv.z + av.w * wv.w;
    }
    s[n] = acc + bs[0];
}

// per-block max (grid-stride) -> partial[blockIdx]
__global__ void block_max(const float* __restrict__ s, float* __restrict__ partial, int N) {
    __shared__ float sm[256];
    float m = -3.4e38f;
    for (int i = blockIdx.x * blockDim.x + threadIdx.x; i < N; i += gridDim.x * blockDim.x)
        m = fmaxf(m, s[i]);
    sm[threadIdx.x] = m;
    __syncthreads();
    for (int off = 128; off > 0; off >>= 1) {
        if (threadIdx.x < off) sm[threadIdx.x] = fmaxf(sm[threadIdx.x], sm[threadIdx.x + off]);
        __syncthreads();
    }
    if (threadIdx.x == 0) partial[blockIdx.x] = sm[0];
}

// Fused softmax-numerator + dot(h,Wo) + per-batch pooling via LDS atomics.
// sums[0] = Z, sums[1..8] = per-batch sum of exp(s-max)*dot(h,Wo), sums[9..16] = counts
__global__ void pool_kernel(const float* __restrict__ s, const float* __restrict__ h,
                            const float* __restrict__ Wo, const int* __restrict__ batch,
                            const float* __restrict__ gmax, float* __restrict__ sums, int N) {
    __shared__ float sZ;
    __shared__ float sU[8];
    __shared__ float sC[8];
    if (threadIdx.x == 0) sZ = 0.0f;
    if (threadIdx.x < 8) { sU[threadIdx.x] = 0.0f; sC[threadIdx.x] = 0.0f; }
    __syncthreads();

    const float m = gmax[0];
    const float4* w = (const float4*)Wo;
    for (int n = blockIdx.x * blockDim.x + threadIdx.x; n < N; n += gridDim.x * blockDim.x) {
        float t = expf(s[n] - m);
        const float4* hp = (const float4*)(h + (size_t)n * 64);
        float acc = 0.0f;
#pragma unroll
        for (int i = 0; i < 16; ++i) {
            float4 hv = hp[i], wv = w[i];
            acc += hv.x * wv.x + hv.y * wv.y + hv.z * wv.z + hv.w * wv.w;
        }
        int b = batch[n];
        atomicAdd(&sZ, t);              // ds_add_f32
        atomicAdd(&sU[b], t * acc);
        atomicAdd(&sC[b], 1.0f);
    }
    __syncthreads();
    if (threadIdx.x == 0) atomicAdd(&sums[0], sZ);
    if (threadIdx.x < 8) {
        atomicAdd(&sums[1 + threadIdx.x], sU[threadIdx.x]);
        atomicAdd(&sums[9 + threadIdx.x], sC[threadIdx.x]);
    }
}

// out[b] = (sumU[b]/Z)/count[b] + bo
__global__ void finalize(const float* __restrict__ sums, const float* __restrict__ bo,
                         float* __restrict__ out, int B) {
    int b = threadIdx.x;
    if (b < B) out[b] = sums[1 + b] / sums[0] / sums[9 + b] + bo[0];
}

// ---------------------------------------------------------------------------
extern "C" void kernel_launch(void* const* d_in, const int* in_sizes, int n_in,
                              void* d_out, int out_size, void* d_ws, size_t ws_size,
                              hipStream_t stream) {
    const float* x    = (const float*)d_in[0];
    const int*   ei   = (const int*)  d_in[1];   // [2,E] int32
    const float* ew   = (const float*)d_in[2];
    const int*   bat  = (const int*)  d_in[3];   // [N]
    const float* app  = (const float*)d_in[4];   // [B,1,64]
    const float* W1   = (const float*)d_in[5];
    const float* b1   = (const float*)d_in[6];
    const float* W2   = (const float*)d_in[7];
    const float* b2   = (const float*)d_in[8];
    const float* Wa   = (const float*)d_in[9];   // [128,64]
    const float* ba   = (const float*)d_in[10];
    const float* Ws   = (const float*)d_in[11];  // [64,1]
    const float* bs   = (const float*)d_in[12];
    const float* Wo   = (const float*)d_in[13];  // [64,1]
    const float* bo   = (const float*)d_in[14];

    const int N  = in_sizes[0] / 64;
    const int E  = in_sizes[2];
    const int B  = in_sizes[4] / 64;
    const int NF = N * 64;

    // workspace layout (floats)
    float* xw   = (float*)d_ws;          // N*64  (also reused as att_hidden)
    float* agg  = xw  + (size_t)NF;      // N*64  (also reused as scores s[N])
    float* h    = agg + (size_t)NF;      // N*64
    float* dis  = h   + (size_t)NF;      // N     (deg -> dis in place)
    float* embW = dis + (size_t)N;       // B*64
    float* red  = embW + 512;            // 257 (partial maxes + gmax at [256])
    float* sums = red + 272;             // 17  (Z, sumU[8], counts[8])

    const int T = 256;
    dim3 blk(T);

    // --- degree / normalization ---
    zero_f<<<(N + T - 1) / T, blk, 0, stream>>>(dis, N);
    zero_f<<<(NF + T - 1) / T, blk, 0, stream>>>(agg, NF);
    zero_f<<<1, 32, 0, stream>>>(sums, 32);
    deg_accum<<<(E + T - 1) / T, blk, 0, stream>>>(ei, ew, dis, E);
    dis_kernel<<<(N + T - 1) / T, blk, 0, stream>>>(dis, N);

    // --- GCN layer 1 ---
    gemm64_wmma<<<N / 16, 128, 0, stream>>>(x, W1, nullptr, nullptr, xw, 0);
    edge_scatter<<<((size_t)E * 16 + T - 1) / T, blk, 0, stream>>>(ei, ew, dis, xw, agg, E);
    combine_relu<<<(NF + T - 1) / T, blk, 0, stream>>>(agg, xw, dis, b1, h, NF);

    // --- GCN layer 2 ---
    zero_f<<<(NF + T - 1) / T, blk, 0, stream>>>(agg, NF);
    gemm64_wmma<<<N / 16, 128, 0, stream>>>(h, W2, nullptr, nullptr, xw, 0);
    edge_scatter<<<((size_t)E * 16 + T - 1) / T, blk, 0, stream>>>(ei, ew, dis, xw, agg, E);
    combine_relu<<<(NF + T - 1) / T, blk, 0, stream>>>(agg, xw, dis, b2, h, NF);

    // --- attention MLP: leaky_relu(h @ Wa[0:64] + embW[batch]) ---
    embw_kernel<<<1, 512, 0, stream>>>(app, Wa, ba, embW);
    gemm64_wmma<<<N / 16, 128, 0, stream>>>(h, Wa, embW, bat, xw, 1);   // att -> xw
    scores_kernel<<<(N + T - 1) / T, blk, 0, stream>>>(xw, Ws, bs, agg, N); // s -> agg

    // --- softmax (global) + weighted pool ---
    block_max<<<256, blk, 0, stream>>>(agg, red, N);
    block_max<<<1, blk, 0, stream>>>(red, red + 256, 256);
    pool_kernel<<<256, blk, 0, stream>>>(agg, h, Wo, bat, red + 256, sums, N);
    finalize<<<1, 32, 0, stream>>>(sums, bo, (float*)d_out, B);
}